// Attention_17102559772682
// MI455X (gfx1250) — compile-verified
//
#include <hip/hip_runtime.h>
#include <hip/hip_bf16.h>

typedef float v2f __attribute__((ext_vector_type(2)));
typedef float v8f __attribute__((ext_vector_type(8)));

#define DIM 768
#define NH 12
#define HD 64
#define BATCH 8
#define HH 56
#define WW 56
#define NN (HH*WW)          // 3136
#define HS 28
#define WS 28
#define MS (HS*WS)          // 784
#define HK 14
#define WK 14
#define MK (HK*WK)          // 196
#define MKP 208             // 196 padded to 13*16

// ---------------------------------------------------------------------------
// Generic GEMM: C[M,N] = (A + A2?) @ W + bias, f32 WMMA 16x16x4.
// One wave computes a 16(M) x 128(N) tile. Grid: (M/16, N/128), block 32.
// A fragment (16x4 f32): lane%16 -> M row, K = k0 + 2*(lane/16) + {x,y}
// B fragment (4x16 f32): lane%16 -> N col, K = k0 + 2*(lane/16) + {x,y}
// D (16x16 f32): lane%16 -> N col, M = vgpr + 8*(lane/16)
// Pointer-increment addressing: no per-iteration integer multiplies, and the
// A2 path is a template parameter so the hot loop is branch-free.
// ---------------------------------------------------------------------------
template <bool HAS_A2>
__global__ __launch_bounds__(32) void gemm_wmma_f32(
    const float* __restrict__ A, const float* __restrict__ A2,
    const float* __restrict__ W, const float* __restrict__ bias,
    float* __restrict__ C, int M, int N, int K)
{
  const int lane = threadIdx.x;
  const int g    = lane >> 4;      // lane group 0/1
  const int ln16 = lane & 15;
  const int m0 = blockIdx.x * 16;
  const int n0 = blockIdx.y * 128;

  v8f acc[8];
#pragma unroll
  for (int t = 0; t < 8; ++t)
#pragma unroll
    for (int i = 0; i < 8; ++i) acc[t][i] = 0.0f;

  const float* ap  = A + (size_t)(m0 + ln16) * K + 2 * g;
  const float* ap2 = HAS_A2 ? (A2 + (size_t)(m0 + ln16) * K + 2 * g) : nullptr;
  const float* wp0 = W + (size_t)(2 * g) * N + n0 + ln16;  // row k0+2g
  const float* wp1 = wp0 + N;                               // row k0+2g+1
  const size_t wstep = 4 * (size_t)N;

  for (int k = 0; k < K; k += 4) {
    v2f a;
    a.x = ap[0];
    a.y = ap[1];
    if (HAS_A2) {
      a.x += ap2[0];
      a.y += ap2[1];
      ap2 += 4;
    }
    ap += 4;
#pragma unroll
    for (int t = 0; t < 8; ++t) {
      v2f b;
      b.x = wp0[t * 16];
      b.y = wp1[t * 16];
      acc[t] = __builtin_amdgcn_wmma_f32_16x16x4_f32(
          false, a, false, b, (short)0, acc[t], false, false);
    }
    wp0 += wstep;
    wp1 += wstep;
  }

#pragma unroll
  for (int t = 0; t < 8; ++t) {
    int col = n0 + t * 16 + ln16;
    float bv = bias[col];
#pragma unroll
    for (int v = 0; v < 8; ++v) {
      int row = m0 + v + 8 * g;
      C[(size_t)row * N + col] = acc[t][v] + bv;
    }
  }
}

// ---------------------------------------------------------------------------
// sr path: depthwise 3x3 stride-2 pad-1 conv over x (as [B,768,56,56]) + bias,
// then LayerNorm over 768 channels (eps 1e-6). Output xs [B,784,768].
// One block (256 thr) per (b, output position); each thread owns 3 channels.
// ---------------------------------------------------------------------------
__global__ __launch_bounds__(256) void sr_conv_ln_kernel(
    const float* __restrict__ x, const float* __restrict__ w,
    const float* __restrict__ cb, const float* __restrict__ gg,
    const float* __restrict__ gb, float* __restrict__ xs)
{
  __shared__ float r1[256];
  __shared__ float r2[256];
  const int tid = threadIdx.x;
  const int blk = blockIdx.x;
  const int b = blk / MS, pos = blk % MS;
  const int oh = pos / WS, ow = pos % WS;

  float vals[3];
  float ls = 0.f, lq = 0.f;
#pragma unroll
  for (int j = 0; j < 3; ++j) {
    int c = tid + j * 256;
    float acc = cb[c];
#pragma unroll
    for (int ky = 0; ky < 3; ++ky) {
      int ih = oh * 2 - 1 + ky;
      if (ih < 0 || ih >= HH) continue;
#pragma unroll
      for (int kx = 0; kx < 3; ++kx) {
        int iw = ow * 2 - 1 + kx;
        if (iw < 0 || iw >= WW) continue;
        acc += x[((size_t)b * NN + ih * WW + iw) * DIM + c] * w[c * 9 + ky * 3 + kx];
      }
    }
    vals[j] = acc; ls += acc; lq += acc * acc;
  }
  r1[tid] = ls; r2[tid] = lq;
  __syncthreads();
  for (int off = 128; off > 0; off >>= 1) {
    if (tid < off) { r1[tid] += r1[tid + off]; r2[tid] += r2[tid + off]; }
    __syncthreads();
  }
  float mean = r1[0] * (1.0f / DIM);
  float var  = r2[0] * (1.0f / DIM) - mean * mean;
  float ri   = rsqrtf(var + 1e-6f);
#pragma unroll
  for (int j = 0; j < 3; ++j) {
    int c = tid + j * 256;
    xs[((size_t)b * MS + pos) * DIM + c] = (vals[j] - mean) * ri * gg[c] + gb[c];
  }
}

// ---------------------------------------------------------------------------
// pool_q: per-head depthwise 3x3 s1 p1 conv (no bias) + LN(64, eps 1e-5).
// q is [B,N,768] (col = h*64+d). Output qp [(b*12+h)*3136+n]*64+d.
// One wave per (b,h,n); each lane owns channels d and d+32.
// ---------------------------------------------------------------------------
__global__ __launch_bounds__(256) void pool_q_ln_kernel(
    const float* __restrict__ q, const float* __restrict__ w,
    const float* __restrict__ gg, const float* __restrict__ gb,
    float* __restrict__ qp)
{
  int wid  = blockIdx.x * 8 + (threadIdx.x >> 5);
  int lane = threadIdx.x & 31;
  int n  = wid % NN;
  int bh = wid / NN;
  int b = bh / NH, h = bh % NH;
  int oh = n / WW, ow = n % WW;
  int d0 = lane, d1 = lane + 32;
  float a0 = 0.f, a1 = 0.f;
#pragma unroll
  for (int ky = 0; ky < 3; ++ky) {
    int ih = oh - 1 + ky;
    if (ih < 0 || ih >= HH) continue;
#pragma unroll
    for (int kx = 0; kx < 3; ++kx) {
      int iw = ow - 1 + kx;
      if (iw < 0 || iw >= WW) continue;
      size_t base = ((size_t)b * NN + ih * WW + iw) * DIM + h * HD;
      a0 += q[base + d0] * w[d0 * 9 + ky * 3 + kx];
      a1 += q[base + d1] * w[d1 * 9 + ky * 3 + kx];
    }
  }
  float s = a0 + a1, sq = a0 * a0 + a1 * a1;
#pragma unroll
  for (int m = 16; m >= 1; m >>= 1) {
    s  += __shfl_xor(s,  m, 32);
    sq += __shfl_xor(sq, m, 32);
  }
  float mean = s * (1.0f / HD);
  float var  = sq * (1.0f / HD) - mean * mean;
  float ri   = rsqrtf(var + 1e-5f);
  size_t o = (size_t)wid * HD;
  qp[o + d0] = (a0 - mean) * ri * gg[d0] + gb[d0];
  qp[o + d1] = (a1 - mean) * ri * gg[d1] + gb[d1];
}

// ---------------------------------------------------------------------------
// pool_k / pool_v: depthwise 3x3 s2 p1 (no bias) on 28x28 -> 14x14 + LN(64).
// kv is [B,784,1536]; k cols 0..767, v cols 768..1535 (col = sel*768+h*64+d).
// grid.y selects k (0) or v (1). One wave per (b,h,m_out).
// ---------------------------------------------------------------------------
__global__ __launch_bounds__(256) void pool_kv_ln_kernel(
    const float* __restrict__ kv,
    const float* __restrict__ wk, const float* __restrict__ wv,
    const float* __restrict__ gk, const float* __restrict__ bk,
    const float* __restrict__ gv, const float* __restrict__ bv,
    float* __restrict__ kp, float* __restrict__ vp)
{
  int sel  = blockIdx.y;
  int wid  = blockIdx.x * 8 + (threadIdx.x >> 5);
  int lane = threadIdx.x & 31;
  int m  = wid % MK;
  int bh = wid / MK;
  int b = bh / NH, h = bh % NH;
  int oh = m / WK, ow = m % WK;
  const float* w  = sel ? wv : wk;
  const float* gg = sel ? gv : gk;
  const float* gb = sel ? bv : bk;
  float* dst = sel ? vp : kp;
  int d0 = lane, d1 = lane + 32;
  float a0 = 0.f, a1 = 0.f;
#pragma unroll
  for (int ky = 0; ky < 3; ++ky) {
    int ih = oh * 2 - 1 + ky;
    if (ih < 0 || ih >= HS) continue;
#pragma unroll
    for (int kx = 0; kx < 3; ++kx) {
      int iw = ow * 2 - 1 + kx;
      if (iw < 0 || iw >= WS) continue;
      size_t base = ((size_t)b * MS + ih * WS + iw) * (2 * DIM) + sel * DIM + h * HD;
      a0 += kv[base + d0] * w[d0 * 9 + ky * 3 + kx];
      a1 += kv[base + d1] * w[d1 * 9 + ky * 3 + kx];
    }
  }
  float s = a0 + a1, sq = a0 * a0 + a1 * a1;
#pragma unroll
  for (int mm = 16; mm >= 1; mm >>= 1) {
    s  += __shfl_xor(s,  mm, 32);
    sq += __shfl_xor(sq, mm, 32);
  }
  float mean = s * (1.0f / HD);
  float var  = sq * (1.0f / HD) - mean * mean;
  float ri   = rsqrtf(var + 1e-5f);
  size_t o = (size_t)wid * HD;
  dst[o + d0] = (a0 - mean) * ri * gg[d0] + gb[d0];
  dst[o + d1] = (a1 - mean) * ri * gg[d1] + gb[d1];
}

// ---------------------------------------------------------------------------
// Identity path: v (from kv buffer) -> depthwise multiplier-4 3x3 s1 p1 conv
// -> PixelShuffle(2) -> LN(768, eps 1e-6).  Output channel c at (oh,ow) pulls
// conv channel oc = c*4 + (oh%2)*2 + (ow%2) sampled at (oh/2, ow/2).
// One block per (b, output token), thread owns 3 channels.
// ---------------------------------------------------------------------------
__global__ __launch_bounds__(256) void up_conv_ln_kernel(
    const float* __restrict__ kv, const float* __restrict__ w,
    const float* __restrict__ cb, const float* __restrict__ gg,
    const float* __restrict__ gb, float* __restrict__ idn)
{
  __shared__ float r1[256];
  __shared__ float r2[256];
  const int tid = threadIdx.x;
  const int blk = blockIdx.x;
  const int b = blk / NN, tok = blk % NN;
  const int oh = tok / WW, ow = tok % WW;
  const int hs = oh >> 1, rr = oh & 1;
  const int ws = ow >> 1, ss = ow & 1;

  float vals[3];
  float ls = 0.f, lq = 0.f;
#pragma unroll
  for (int j = 0; j < 3; ++j) {
    int c = tid + j * 256;
    int oc = c * 4 + rr * 2 + ss;
    float acc = cb[oc];
#pragma unroll
    for (int ky = 0; ky < 3; ++ky) {
      int ih = hs - 1 + ky;
      if (ih < 0 || ih >= HS) continue;
#pragma unroll
      for (int kx = 0; kx < 3; ++kx) {
        int iw = ws - 1 + kx;
        if (iw < 0 || iw >= WS) continue;
        acc += kv[((size_t)b * MS + ih * WS + iw) * (2 * DIM) + DIM + c] * w[oc * 9 + ky * 3 + kx];
      }
    }
    vals[j] = acc; ls += acc; lq += acc * acc;
  }
  r1[tid] = ls; r2[tid] = lq;
  __syncthreads();
  for (int off = 128; off > 0; off >>= 1) {
    if (tid < off) { r1[tid] += r1[tid + off]; r2[tid] += r2[tid + off]; }
    __syncthreads();
  }
  float mean = r1[0] * (1.0f / DIM);
  float var  = r2[0] * (1.0f / DIM) - mean * mean;
  float ri   = rsqrtf(var + 1e-6f);
#pragma unroll
  for (int j = 0; j < 3; ++j) {
    int c = tid + j * 256;
    idn[((size_t)b * NN + tok) * DIM + c] = (vals[j] - mean) * ri * gg[c] + gb[c];
  }
}

// ---------------------------------------------------------------------------
// Fused attention: one wave per (b*12+h, 16 query rows).
// Phase 1: S[16,208] = q @ kp^T via WMMA (13 N-tiles, K=64), scale+mask -> LDS
// Phase 2: row softmax in LDS (unnormalized exp kept; row sums saved)
// Phase 3: O[16,64] = P @ vp via WMMA (K=208 incl. zeroed padding), /rowsum
// ---------------------------------------------------------------------------
__global__ __launch_bounds__(32) void attn_kernel(
    const float* __restrict__ qp, const float* __restrict__ kp,
    const float* __restrict__ vp, float* __restrict__ out)
{
  __shared__ float sP[16 * MKP];
  __shared__ float sSum[16];
  const int lane = threadIdx.x;
  const int g    = lane >> 4;
  const int ln16 = lane & 15;
  const int q0 = blockIdx.x * 16;
  const int bh = blockIdx.y;
  const int b = bh / NH, h = bh % NH;
  const float* qb = qp + (size_t)bh * NN * HD;
  const float* kb = kp + (size_t)bh * MK * HD;
  const float* vb = vp + (size_t)bh * MK * HD;

  // Phase 1: S = q @ kp^T
  v8f acc[13];
#pragma unroll
  for (int t = 0; t < 13; ++t)
#pragma unroll
    for (int i = 0; i < 8; ++i) acc[t][i] = 0.0f;

  {
    const float* qr = qb + (size_t)(q0 + ln16) * HD + 2 * g;
    const float* kr = kb + (size_t)ln16 * HD + 2 * g;
    for (int k = 0; k < HD; k += 4) {
      v2f a;
      a.x = qr[0]; a.y = qr[1];
#pragma unroll
      for (int t = 0; t < 13; ++t) {
        v2f bf;  // B[d][n] = kp[n*64 + d]
        bf.x = kr[t * 16 * HD];
        bf.y = kr[t * 16 * HD + 1];
        acc[t] = __builtin_amdgcn_wmma_f32_16x16x4_f32(
            false, a, false, bf, (short)0, acc[t], false, false);
      }
      qr += 4;
      kr += 4;
    }
  }

  const float scale = 0.125f;  // 64^-0.5
#pragma unroll
  for (int t = 0; t < 13; ++t) {
    int col = t * 16 + ln16;
#pragma unroll
    for (int v = 0; v < 8; ++v) {
      int row = v + 8 * g;
      sP[row * MKP + col] = (col < MK) ? acc[t][v] * scale : -3.0e38f;
    }
  }
  __syncthreads();

  // Phase 2: softmax rows (row = ln16; lane groups split cols by parity)
  {
    int row = ln16;
    float m = -3.0e38f;
    for (int c = g; c < MKP; c += 2) m = fmaxf(m, sP[row * MKP + c]);
    m = fmaxf(m, __shfl_xor(m, 16, 32));
    float s = 0.f;
    for (int c = g; c < MKP; c += 2) {
      float e = __expf(sP[row * MKP + c] - m);
      sP[row * MKP + c] = e;
      s += e;
    }
    s += __shfl_xor(s, 16, 32);
    if (g == 0) sSum[row] = s;
  }
  __syncthreads();

  // Phase 3: O = P @ vp   (padded P cols are exactly 0 -> padded vp reads moot)
  v8f oc[4];
#pragma unroll
  for (int t = 0; t < 4; ++t)
#pragma unroll
    for (int i = 0; i < 8; ++i) oc[t][i] = 0.0f;

  {
    const float* pr = sP + ln16 * MKP + 2 * g;
    const float* vr = vb + (size_t)(2 * g) * HD + ln16;
    for (int k = 0; k < MKP; k += 4) {
      v2f a;
      a.x = pr[0];
      a.y = pr[1];
#pragma unroll
      for (int t = 0; t < 4; ++t) {
        v2f bf;  // B[n][e] = vp[n*64 + e]
        bf.x = vr[t * 16];
        bf.y = vr[t * 16 + HD];
        oc[t] = __builtin_amdgcn_wmma_f32_16x16x4_f32(
            false, a, false, bf, (short)0, oc[t], false, false);
      }
      pr += 4;
      vr += 4 * HD;
    }
  }

#pragma unroll
  for (int t = 0; t < 4; ++t) {
    int e = t * 16 + ln16;
#pragma unroll
    for (int v = 0; v < 8; ++v) {
      int row = v + 8 * g;
      float rs = sSum[row];
      out[((size_t)b * NN + q0 + row) * DIM + h * HD + e] = oc[t][v] / rs;
    }
  }
}

// ---------------------------------------------------------------------------
extern "C" void kernel_launch(void* const* d_in, const int* in_sizes, int n_in,
                              void* d_out, int out_size, void* d_ws, size_t ws_size,
                              hipStream_t stream) {
  const float* x      = (const float*)d_in[0];
  const float* q_w    = (const float*)d_in[1];
  const float* q_b    = (const float*)d_in[2];
  const float* kv_w   = (const float*)d_in[3];
  const float* kv_b   = (const float*)d_in[4];
  const float* sr_w   = (const float*)d_in[5];
  const float* sr_b   = (const float*)d_in[6];
  const float* srn_g  = (const float*)d_in[7];
  const float* srn_b  = (const float*)d_in[8];
  const float* up_w   = (const float*)d_in[9];
  const float* up_b   = (const float*)d_in[10];
  const float* upn_g  = (const float*)d_in[11];
  const float* upn_b  = (const float*)d_in[12];
  const float* proj_w = (const float*)d_in[13];
  const float* proj_b = (const float*)d_in[14];
  const float* pq_w   = (const float*)d_in[15];
  const float* pk_w   = (const float*)d_in[16];
  const float* pv_w   = (const float*)d_in[17];
  const float* nq_g   = (const float*)d_in[18];
  const float* nq_b   = (const float*)d_in[19];
  const float* nk_g   = (const float*)d_in[20];
  const float* nk_b   = (const float*)d_in[21];
  const float* nv_g   = (const float*)d_in[22];
  const float* nv_b   = (const float*)d_in[23];

  float* ws = (float*)d_ws;
  const size_t SZ_BND = (size_t)BATCH * NN * DIM;        // 19,267,584
  const size_t SZ_XS  = (size_t)BATCH * MS * DIM;        //  4,816,896
  const size_t SZ_KV  = (size_t)BATCH * MS * 2 * DIM;    //  9,633,792
  const size_t SZ_KP  = (size_t)BATCH * NH * MK * HD + 1024;  // +4KB tail pad

  float* qbuf = ws;                    // q proj; reused as attention output
  float* qpb  = qbuf + SZ_BND;         // pooled+LN q
  float* xsb  = qpb + SZ_BND;          // sr-reduced x
  float* kvb  = xsb + SZ_XS;           // kv proj
  float* kpb  = kvb + SZ_KV;           // pooled+LN k (padded tail)
  float* vpb  = kpb + SZ_KP;           // pooled+LN v (padded tail)
  float* idnb = vpb + SZ_KP;           // identity path
  float* outb = qbuf;                  // attention out (q consumed by then)

  const int Mtok = BATCH * NN;         // 25088
  const int Mred = BATCH * MS;         // 6272

  // 1. q = x @ q_w + q_b
  gemm_wmma_f32<false><<<dim3(Mtok / 16, DIM / 128), 32, 0, stream>>>(
      x, nullptr, q_w, q_b, qbuf, Mtok, DIM, DIM);

  // 2. spatial reduction conv + LN(768)
  sr_conv_ln_kernel<<<BATCH * MS, 256, 0, stream>>>(
      x, sr_w, sr_b, srn_g, srn_b, xsb);

  // 3. kv = xs @ kv_w + kv_b
  gemm_wmma_f32<false><<<dim3(Mred / 16, (2 * DIM) / 128), 32, 0, stream>>>(
      xsb, nullptr, kv_w, kv_b, kvb, Mred, 2 * DIM, DIM);

  // 4. pool_q + LN(64)
  pool_q_ln_kernel<<<(BATCH * NH * NN) / 8, 256, 0, stream>>>(
      qbuf, pq_w, nq_g, nq_b, qpb);

  // 5. pool_k / pool_v + LN(64)
  pool_kv_ln_kernel<<<dim3((BATCH * NH * MK) / 8, 2), 256, 0, stream>>>(
      kvb, pk_w, pv_w, nk_g, nk_b, nv_g, nv_b, kpb, vpb);

  // 6. identity path conv + pixel shuffle + LN(768)
  up_conv_ln_kernel<<<BATCH * NN, 256, 0, stream>>>(
      kvb, up_w, up_b, upn_g, upn_b, idnb);

  // 7. attention core (writes into qbuf region)
  attn_kernel<<<dim3(NN / 16, BATCH * NH), 32, 0, stream>>>(
      qpb, kpb, vpb, outb);

  // 8. final: (out + idn) @ proj_w + proj_b -> d_out
  gemm_wmma_f32<true><<<dim3(Mtok / 16, DIM / 128), 32, 0, stream>>>(
      outb, idnb, proj_w, proj_b, (float*)d_out, Mtok, DIM, DIM);
}